// topo_loss_55113020342426
// MI455X (gfx1250) — compile-verified
//
#include <hip/hip_runtime.h>

#define HH 64
#define WW 64
#define NN 4096          // HH*WW
#define NPROB 16         // 8 images x {sublevel, superlevel}
#define THREADS 256

typedef __attribute__((ext_vector_type(2))) float v2f;
typedef __attribute__((ext_vector_type(8))) float v8f;

// ---------------------------------------------------------------------------
// Kernel 1: one block per persistence problem. Whole problem lives in LDS:
//   sval [4096]f  sorted filtration values
//   sidx [4096]i  argsort permutation (order)
//   rank [4096]i  inverse permutation
//   parent[4096]i union-find forest
//   death[4096]f  death values -> persistence -> sorted persistence
// ---------------------------------------------------------------------------
extern "C" __global__ __launch_bounds__(THREADS)
void persist_kernel(const float* __restrict__ pred,
                    const float* __restrict__ gt,
                    float* __restrict__ ws_pers) {
  extern __shared__ char smem[];
  float* sval   = (float*)(smem);
  int*   sidx   = (int*)  (smem + NN * 4);
  int*   rank   = (int*)  (smem + NN * 8);
  int*   parent = (int*)  (smem + NN * 12);
  float* death  = (float*)(smem + NN * 16);

  const int b   = blockIdx.x;      // problem id
  const int img = b >> 1;          // 0..7 (0-3 pred, 4-7 gt)
  const int dir = b & 1;           // 0 = sublevel(flat), 1 = sublevel(-flat)
  const float* src = (img < 4) ? (pred + (size_t)(img * 2 + 1) * NN)
                               : (gt   + (size_t)((img - 4) * 2 + 1) * NN);
  const int tid = threadIdx.x;

  // ---- load image (negated for superlevel problems) ----
  for (int i = tid; i < NN; i += THREADS) {
    __builtin_prefetch(src + i + THREADS, 0, 3);   // global_prefetch_b8
    float v = src[i];
    sval[i] = dir ? -v : v;
    sidx[i] = i;
  }
  __syncthreads();

  // ---- bitonic argsort: (value asc, index asc) == stable ascending sort ----
  for (int size = 2; size <= NN; size <<= 1) {
    for (int stride = size >> 1; stride > 0; stride >>= 1) {
      for (int t = tid; t < NN / 2; t += THREADS) {
        int i = ((t & ~(stride - 1)) << 1) | (t & (stride - 1));
        int j = i | stride;
        bool asc = ((i & size) == 0);
        float va = sval[i], vb = sval[j];
        int   ia = sidx[i], ib = sidx[j];
        bool swp = asc ? ((va > vb) || (va == vb && ia > ib))
                       : ((va < vb) || (va == vb && ia < ib));
        if (swp) { sval[i] = vb; sval[j] = va; sidx[i] = ib; sidx[j] = ia; }
      }
      __syncthreads();
    }
  }

  // ---- rank (inverse permutation), forest init ----
  for (int k = tid; k < NN; k += THREADS) { rank[sidx[k]] = k; parent[k] = k; }
  __syncthreads();
  // deaths initialized to the filtration value (unpaired slots -> pers 0)
  for (int i = tid; i < NN; i += THREADS) death[i] = sval[rank[i]];
  __syncthreads();

  // ---- sequential elder-rule union-find sweep (single lane) ----
  if (tid == 0) {
    for (int k = 0; k < NN; ++k) {
      int   p = sidx[k];
      float v = sval[k];
      int y = p >> 6, x = p & 63;
      int nbs[4];
      nbs[0] = (y > 0)      ? p - WW : -1;   // up
      nbs[1] = (y < HH - 1) ? p + WW : -1;   // down
      nbs[2] = (x > 0)      ? p - 1  : -1;   // left
      nbs[3] = (x < WW - 1) ? p + 1  : -1;   // right
      for (int t = 0; t < 4; ++t) {
        int nb = nbs[t];
        if (nb < 0 || rank[nb] >= k) continue;
        // find with path halving (roots are preserved -> essential test intact)
        int rp = p;
        while (parent[rp] != rp) { parent[rp] = parent[parent[rp]]; rp = parent[rp]; }
        int rn = nb;
        while (parent[rn] != rn) { parent[rn] = parent[parent[rn]]; rn = parent[rn]; }
        int elder, younger;
        if (rank[rn] < rank[rp]) { elder = rn; younger = rp; }
        else                     { elder = rp; younger = rn; }
        if (rp != rn) death[younger] = v;     // younger component dies at v
        parent[younger] = elder;
        parent[p]       = elder;
      }
    }
  }
  __syncthreads();

  // ---- persistence values (essential: max-flat for p0, 0 for p1) ----
  float vmax = sval[NN - 1];
  for (int i = tid; i < NN; i += THREADS) {
    float flati = sval[rank[i]];
    float pers  = death[i] - flati;
    if (parent[i] == i) pers = dir ? 0.0f : (vmax - flati);
    death[i] = pers;
  }
  __syncthreads();

  // ---- bitonic sort of persistence values (ascending, values only) ----
  for (int size = 2; size <= NN; size <<= 1) {
    for (int stride = size >> 1; stride > 0; stride >>= 1) {
      for (int t = tid; t < NN / 2; t += THREADS) {
        int i = ((t & ~(stride - 1)) << 1) | (t & (stride - 1));
        int j = i | stride;
        bool asc = ((i & size) == 0);
        float va = death[i], vb = death[j];
        bool swp = asc ? (va > vb) : (va < vb);
        if (swp) { death[i] = vb; death[j] = va; }
      }
      __syncthreads();
    }
  }

  for (int k = tid; k < NN; k += THREADS)
    ws_pers[(size_t)b * NN + k] = death[k];
}

// ---------------------------------------------------------------------------
// Kernel 2: W1 = 0.5 * sum |sort(pa) - sort(pb)| per diagram pair, reduced on
// the matrix pipe: V_WMMA_F32_16X16X4_F32 with B = all-ones -> f32 row sums.
// One wave (32 lanes, EXEC all ones) per pair; 64 chained WMMAs cover 4096
// diffs; D entries replicate each row-sum across 16 columns -> divide by 16.
// ---------------------------------------------------------------------------
extern "C" __global__ __launch_bounds__(32)
void w1_kernel(const float* __restrict__ ws_pers, float* __restrict__ out) {
  const int q = blockIdx.x;          // pair 0..7
  const int i = q & 3;               // image index
  const int d = q >> 2;              // 0 = p0, 1 = p1
  const float* pa = ws_pers + (size_t)(2 * i + d) * NN;        // pred problem
  const float* pb = ws_pers + (size_t)(2 * (4 + i) + d) * NN;  // gt problem

  const int lane = threadIdx.x;
  const int m    = lane & 15;          // A-matrix row
  const int kofs = (lane >> 4) * 2;    // lanes 16-31 hold K=2,3

  v8f acc = {};                        // C/D accumulator (16x16 f32)
  v2f ones; ones.x = 1.0f; ones.y = 1.0f;   // B = all ones (layout-invariant)

  for (int chunk = 0; chunk < NN / 64; ++chunk) {
    int base = chunk * 64 + m * 4 + kofs;
    v2f a;
    a.x = fabsf(pa[base]     - pb[base]);
    a.y = fabsf(pa[base + 1] - pb[base + 1]);
    // D = A(16x4) * ones(4x16) + C   -> accumulates |diff| row sums in f32
    acc = __builtin_amdgcn_wmma_f32_16x16x4_f32(
        false, a, false, ones, (short)0, acc, false, false);
  }

  float s = acc[0] + acc[1] + acc[2] + acc[3] + acc[4] + acc[5] + acc[6] + acc[7];
  for (int off = 16; off > 0; off >>= 1) s += __shfl_xor(s, off, 32);
  if (lane == 0) atomicAdd(out, 0.5f * (s * (1.0f / 16.0f)));
}

extern "C" __global__ void zero_out_kernel(float* out) {
  if (threadIdx.x == 0 && blockIdx.x == 0) out[0] = 0.0f;
}

// ---------------------------------------------------------------------------
extern "C" void kernel_launch(void* const* d_in, const int* in_sizes, int n_in,
                              void* d_out, int out_size, void* d_ws, size_t ws_size,
                              hipStream_t stream) {
  const float* pred = (const float*)d_in[0];   // (4,2,64,64) f32
  const float* gt   = (const float*)d_in[1];   // (4,2,64,64) f32
  float* out        = (float*)d_out;           // scalar f32
  float* ws_pers    = (float*)d_ws;            // 16 * 4096 f32 = 256 KB

  zero_out_kernel<<<1, 1, 0, stream>>>(out);

  const size_t lds_bytes = (size_t)NN * 4 * 5;  // 80 KB dynamic LDS / block
  persist_kernel<<<NPROB, THREADS, lds_bytes, stream>>>(pred, gt, ws_pers);

  w1_kernel<<<8, 32, 0, stream>>>(ws_pers, out);
}